// QECCEqualModel_51951924412729
// MI455X (gfx1250) — compile-verified
//
// QECC overlap loss — CDNA5 (gfx1250) fused-gate implementation.
//
// Roofline: 5.4 GFLOP vs 23.3 TB/s HBM -> memory bound. Naive per-gate passes
// = ~5.9 GB traffic (~250us). Here: gates fused into six 16x16 group
// unitaries, applied as V_WMMA_F32_16X16X4_F32 chains on LDS-resident tiles,
// 3 passes over the 128MB state + fused overlap reduction => ~0.9 GB (~38us).
//
// Workspace requirement: 8192*4 + 4*DIM*4 bytes  (~128.03 MB).

#include <hip/hip_runtime.h>

#define NQ  23
#define DIM (1u << NQ)   // 8388608

typedef float v2f __attribute__((ext_vector_type(2)));
typedef float v8f __attribute__((ext_vector_type(8)));

// D(16x16,f32) = A(16x4,f32) * B(4x16,f32) + C   — chained over K to build 16x16x16.
static __device__ __forceinline__ v8f wmma_f32_k4(v2f a, v2f b, v8f c) {
  return __builtin_amdgcn_wmma_f32_16x16x4_f32(
      /*neg_a=*/false, a, /*neg_b=*/false, b,
      /*c_mod=*/(short)0, c, /*reuse_a=*/false, /*reuse_b=*/false);
}

// Address of amplitude inside a 4096-element LDS tile for a group whose 4 index
// bits sit at shift GS; B8 = 8-bit batch id (the other 8 local bits), k = group idx.
template<int GS>
static __device__ __forceinline__ int grp_addr(int B8, int k) {
  int lo = B8 & ((1 << GS) - 1);
  int hi = B8 >> GS;
  return (hi << (GS + 4)) | (k << GS) | lo;
}

// One wave applies one fused 16x16 complex unitary (Wg: 256 re + 256 im floats)
// to its 2 tiles (16 batches x 16 group elements each) of the 4096-amp LDS block.
// WMMA VGPR layouts per CDNA5 ISA 7.12.2 (A 16x4, B 4x16, C/D 16x16 f32).
template<int GS>
static __device__ __forceinline__ void wave_apply_group(
    float* __restrict__ lr, float* __restrict__ li,
    const float* __restrict__ Wg, int lane, int waveId)
{
  const float* Wr = Wg;
  const float* Wi = Wg + 256;
  const int n    = lane & 15;
  const int koff = (lane >> 4) << 1;     // lanes 16..31 hold K+2
  v2f Br[4], Bi[4], Bn[4];
#pragma unroll
  for (int c = 0; c < 4; ++c) {
    int k0 = 4 * c + koff;
    Br[c].x = Wr[k0 * 16 + n];  Br[c].y = Wr[(k0 + 1) * 16 + n];
    Bi[c].x = Wi[k0 * 16 + n];  Bi[c].y = Wi[(k0 + 1) * 16 + n];
    Bn[c].x = -Bi[c].x;         Bn[c].y = -Bi[c].y;
  }
  const int M = lane & 15;
#pragma unroll
  for (int tt = 0; tt < 2; ++tt) {
    const int t  = waveId * 2 + tt;      // tile 0..15; disjoint batch ranges
    const int B8 = t * 16 + M;
    v2f Ar[4], Ai[4];
#pragma unroll
    for (int c = 0; c < 4; ++c) {
      int k0 = 4 * c + koff;
      int a0 = grp_addr<GS>(B8, k0);
      int a1 = grp_addr<GS>(B8, k0 + 1);
      Ar[c].x = lr[a0]; Ar[c].y = lr[a1];
      Ai[c].x = li[a0]; Ai[c].y = li[a1];
    }
    v8f Dr = {}; v8f Di = {};
#pragma unroll
    for (int c = 0; c < 4; ++c) {
      Dr = wmma_f32_k4(Ar[c], Br[c], Dr);   // + Ar*Wr
      Dr = wmma_f32_k4(Ai[c], Bn[c], Dr);   // - Ai*Wi
      Di = wmma_f32_k4(Ar[c], Bi[c], Di);   // + Ar*Wi
      Di = wmma_f32_k4(Ai[c], Br[c], Di);   // + Ai*Wr
    }
    const int Mw = (lane >> 4) << 3;        // D: lanes 16..31 -> rows +8
    const int y  = lane & 15;
#pragma unroll
    for (int v = 0; v < 8; ++v) {
      int aw = grp_addr<GS>(t * 16 + v + Mw, y);
      lr[aw] = Dr[v];
      li[aw] = Di[v];
    }
  }
}

// Build six fused group unitaries. Group g<5 covers global bits 4g..4g+3;
// group 5 covers bits 20..22 with identity in slot 3 (=> block-diag 16x16).
// Reference qubit ind0 acts on global bit (22-ind0); U[b][d] at ind0*4+b*2+d.
__global__ void build_groups(const float* __restrict__ ur,
                             const float* __restrict__ ui,
                             float* __restrict__ W)
{
  int t = threadIdx.x;           // 0..255
  int x = t >> 4, y = t & 15;
  for (int g = 0; g < 6; ++g) {
    float pr = 1.f, pi = 0.f;
#pragma unroll
    for (int k = 0; k < 4; ++k) {
      int xb = (x >> k) & 1, yb = (y >> k) & 1;
      float mr, mi;
      if (g == 5 && k == 3) {            // identity slot (block selector)
        mr = (xb == yb) ? 1.f : 0.f; mi = 0.f;
      } else {
        int p   = (g < 5) ? (4 * g + k) : (20 + k);
        int ind = 22 - p;
        int off = ind * 4 + xb * 2 + yb;
        mr = ur[off]; mi = ui[off];
      }
      float nr = pr * mr - pi * mi;
      float ni = pr * mi + pi * mr;
      pr = nr; pi = ni;
    }
    W[g * 512 + x * 16 + y]       = pr;
    W[g * 512 + 256 + x * 16 + y] = pi;
  }
}

// Pass 1: contiguous 4096-amp blocks; apply groups on bits 0-3, 4-7, 8-11.
__global__ void pass1(const float* __restrict__ c0r, const float* __restrict__ c0i,
                      const float* __restrict__ W,
                      float* __restrict__ sr, float* __restrict__ si)
{
  __shared__ float lr[4096], li[4096];
  const int tid = threadIdx.x;
  const int w   = blockIdx.x;                 // 2 states * 2048 blocks
  const int s   = w >> 11, blk = w & 2047;
  const size_t base = (size_t)s * DIM + (size_t)blk * 4096;
#pragma unroll
  for (int j = 0; j < 4; ++j) {
    ((float4*)lr)[j * 256 + tid] = ((const float4*)(c0r + base))[j * 256 + tid];
    ((float4*)li)[j * 256 + tid] = ((const float4*)(c0i + base))[j * 256 + tid];
  }
  __syncthreads();
  const int lane = tid & 31, waveId = tid >> 5;
  wave_apply_group<0>(lr, li, W + 0 * 512, lane, waveId); __syncthreads();
  wave_apply_group<4>(lr, li, W + 1 * 512, lane, waveId); __syncthreads();
  wave_apply_group<8>(lr, li, W + 2 * 512, lane, waveId); __syncthreads();
#pragma unroll
  for (int j = 0; j < 4; ++j) {
    ((float4*)(sr + base))[j * 256 + tid] = ((float4*)lr)[j * 256 + tid];
    ((float4*)(si + base))[j * 256 + tid] = ((float4*)li)[j * 256 + tid];
  }
}

// Pass 2: gather bits 12-19 (H8) x 16 contiguous low amps; apply groups 3,4 in place.
// Local index l = (H8<<4)|in4 => groups sit at local shifts 4 and 8.
__global__ void pass2(const float* __restrict__ W,
                      float* __restrict__ sr, float* __restrict__ si)
{
  __shared__ float lr[4096], li[4096];
  const int tid = threadIdx.x;
  const int w   = blockIdx.x;                 // 2 * 8(h3) * 256(f8)
  const int s   = w >> 11;
  const int h3  = (w >> 8) & 7;
  const int f8  = w & 255;
  const size_t sbase = (size_t)s * DIM + ((size_t)h3 << 20) + ((size_t)f8 << 4);
#pragma unroll
  for (int j = 0; j < 4; ++j) {
    size_t ga = sbase + ((size_t)tid << 12) + (size_t)(j * 4);
    ((float4*)lr)[tid * 4 + j] = *(const float4*)(sr + ga);
    ((float4*)li)[tid * 4 + j] = *(const float4*)(si + ga);
  }
  __syncthreads();
  const int lane = tid & 31, waveId = tid >> 5;
  wave_apply_group<4>(lr, li, W + 3 * 512, lane, waveId); __syncthreads();
  wave_apply_group<8>(lr, li, W + 4 * 512, lane, waveId); __syncthreads();
#pragma unroll
  for (int j = 0; j < 4; ++j) {
    size_t ga = sbase + ((size_t)tid << 12) + (size_t)(j * 4);
    *(float4*)(sr + ga) = ((float4*)lr)[tid * 4 + j];
    *(float4*)(si + ga) = ((float4*)li)[tid * 4 + j];
  }
}

// Pass 3: gather bits 20-22 (h3) x 512 contiguous amps (in9); apply group 5 as a
// block-diagonal 16x16 (two 8-groups per WMMA row), then fuse the overlap
// reduction against code1 (never writes state back to HBM).
__global__ void pass3(const float* __restrict__ W,
                      const float* __restrict__ sr, const float* __restrict__ si,
                      const float* __restrict__ c1r, const float* __restrict__ c1i,
                      float* __restrict__ ov)
{
  __shared__ float lr[4096], li[4096];
  __shared__ float red[4][256];
  const int tid = threadIdx.x;
  const int w   = blockIdx.x;                 // 2 * 2048(F11 = bits 9..19)
  const int s   = w >> 11;
  const int F11 = w & 2047;
  const size_t sbase = (size_t)s * DIM + ((size_t)F11 << 9);
#pragma unroll
  for (int j = 0; j < 4; ++j) {
    int l4 = tid * 4 + j, l = l4 * 4;
    int h3 = l >> 9, in9 = l & 511;
    size_t ga = sbase + ((size_t)h3 << 20) + (size_t)in9;
    ((float4*)lr)[l4] = *(const float4*)(sr + ga);
    ((float4*)li)[l4] = *(const float4*)(si + ga);
  }
  {
    // Warm L2/WGP$ for the code1 operand used after the transform.
    int l = tid * 16;
    int h3 = l >> 9, in9 = l & 511;
    size_t gi = ((size_t)h3 << 20) + ((size_t)F11 << 9) + (size_t)in9;
    __builtin_prefetch(c1r + gi, 0, 0);
    __builtin_prefetch(c1i + gi, 0, 0);
    __builtin_prefetch(c1r + DIM + gi, 0, 0);
    __builtin_prefetch(c1i + DIM + gi, 0, 0);
  }
  __syncthreads();
  const int lane = tid & 31, waveId = tid >> 5;
  {
    const float* Wr = W + 5 * 512;
    const float* Wi = Wr + 256;
    const int n    = lane & 15;
    const int koff = (lane >> 4) << 1;
    v2f Br[4], Bi[4], Bn[4];
#pragma unroll
    for (int c = 0; c < 4; ++c) {
      int k0 = 4 * c + koff;
      Br[c].x = Wr[k0 * 16 + n];  Br[c].y = Wr[(k0 + 1) * 16 + n];
      Bi[c].x = Wi[k0 * 16 + n];  Bi[c].y = Wi[(k0 + 1) * 16 + n];
      Bn[c].x = -Bi[c].x;         Bn[c].y = -Bi[c].y;
    }
    const int M = lane & 15;
#pragma unroll
    for (int tt = 0; tt < 2; ++tt) {
      const int t   = waveId * 2 + tt;     // 0..15
      const int row = t * 16 + M;          // row packs 8-groups 2*row, 2*row+1
      v2f Ar[4], Ai[4];
#pragma unroll
      for (int c = 0; c < 4; ++c) {
        int k0 = 4 * c + koff, k1 = k0 + 1;
        int a0 = ((k0 & 7) << 9) | (2 * row + (k0 >> 3));
        int a1 = ((k1 & 7) << 9) | (2 * row + (k1 >> 3));
        Ar[c].x = lr[a0]; Ar[c].y = lr[a1];
        Ai[c].x = li[a0]; Ai[c].y = li[a1];
      }
      v8f Dr = {}; v8f Di = {};
#pragma unroll
      for (int c = 0; c < 4; ++c) {
        Dr = wmma_f32_k4(Ar[c], Br[c], Dr);
        Dr = wmma_f32_k4(Ai[c], Bn[c], Dr);
        Di = wmma_f32_k4(Ar[c], Bi[c], Di);
        Di = wmma_f32_k4(Ai[c], Br[c], Di);
      }
      const int Mw = (lane >> 4) << 3;
      const int y  = lane & 15;
#pragma unroll
      for (int v = 0; v < 8; ++v) {
        int roww = t * 16 + v + Mw;
        int aw = ((y & 7) << 9) | (2 * roww + (y >> 3));
        lr[aw] = Dr[v];
        li[aw] = Di[v];
      }
    }
  }
  __syncthreads();
  // overlap[s,b] += sum conj(code0')*code1
  float a0r = 0.f, a0i = 0.f, a1r = 0.f, a1i = 0.f;
#pragma unroll
  for (int e = 0; e < 16; ++e) {
    int l = tid * 16 + e;
    int h3 = l >> 9, in9 = l & 511;
    size_t gi = ((size_t)h3 << 20) + ((size_t)F11 << 9) + (size_t)in9;
    float zr = lr[l], zi = li[l];
    float b0r = c1r[gi],        b0i = c1i[gi];
    float b1r = c1r[DIM + gi],  b1i = c1i[DIM + gi];
    a0r += zr * b0r + zi * b0i;  a0i += zr * b0i - zi * b0r;
    a1r += zr * b1r + zi * b1i;  a1i += zr * b1i - zi * b1r;
  }
  red[0][tid] = a0r; red[1][tid] = a0i; red[2][tid] = a1r; red[3][tid] = a1i;
  __syncthreads();
  for (int st = 128; st > 0; st >>= 1) {
    if (tid < st) {
#pragma unroll
      for (int j = 0; j < 4; ++j) red[j][tid] += red[j][tid + st];
    }
    __syncthreads();
  }
  if (tid == 0) {
    atomicAdd(&ov[(s * 2 + 0) * 2 + 0], red[0][0]);
    atomicAdd(&ov[(s * 2 + 0) * 2 + 1], red[1][0]);
    atomicAdd(&ov[(s * 2 + 1) * 2 + 0], red[2][0]);
    atomicAdd(&ov[(s * 2 + 1) * 2 + 1], red[3][0]);
  }
}

__global__ void init_ov(float* __restrict__ ov) {
  if (threadIdx.x < 8) ov[threadIdx.x] = 0.f;
}

__global__ void finalize_k(const float* __restrict__ ov, float* __restrict__ out) {
  if (threadIdx.x == 0 && blockIdx.x == 0) {
    float acc = 0.f;
#pragma unroll
    for (int j = 0; j < 4; ++j) {
      float re = ov[2 * j], im = ov[2 * j + 1];
      acc += re * re + im * im;
    }
    out[0] = 2.0f - acc;    // loss = n0 - |overlap|^2
  }
}

extern "C" void kernel_launch(void* const* d_in, const int* in_sizes, int n_in,
                              void* d_out, int out_size, void* d_ws, size_t ws_size,
                              hipStream_t stream) {
  (void)in_sizes; (void)n_in; (void)out_size; (void)ws_size;
  const float* c0r = (const float*)d_in[0];
  const float* c0i = (const float*)d_in[1];
  const float* c1r = (const float*)d_in[2];
  const float* c1i = (const float*)d_in[3];
  const float* ur  = (const float*)d_in[4];
  const float* ui  = (const float*)d_in[5];

  float* wsf = (float*)d_ws;
  float* W   = wsf;                          // 6*512 floats (fused unitaries)
  float* ov  = wsf + 4096;                   // 8 floats (2x2 complex overlap)
  float* sr  = wsf + 8192;                   // 2*DIM floats (state real)
  float* si  = sr + (size_t)2 * DIM;         // 2*DIM floats (state imag)

  init_ov    <<<1,    32,  0, stream>>>(ov);
  build_groups<<<1,   256, 0, stream>>>(ur, ui, W);
  pass1      <<<4096, 256, 0, stream>>>(c0r, c0i, W, sr, si);
  pass2      <<<4096, 256, 0, stream>>>(W, sr, si);
  pass3      <<<4096, 256, 0, stream>>>(W, sr, si, c1r, c1i, ov);
  finalize_k <<<1,    64,  0, stream>>>(ov, (float*)d_out);
}